// JointLearning1_55448027791637
// MI455X (gfx1250) — compile-verified
//
#include <hip/hip_runtime.h>
#include <hip/hip_bf16.h>

typedef __attribute__((ext_vector_type(2))) float v2f;
typedef __attribute__((ext_vector_type(8))) float v8f;

#define DIM 128
#define ROWP 132               // padded LDS row stride (floats): stride%64 = 4 -> conflict-free column access
#define WAVES_PER_BLOCK 4
#define THREADS (WAVES_PER_BLOCK * 32)

// ---------------------------------------------------------------------------
// Kernel 0: zero the 3 scalar accumulators (ws is poisoned, never re-zeroed)
// ---------------------------------------------------------------------------
__global__ void jl_zero_scalars(float* scal) {
    if (threadIdx.x < 4) scal[threadIdx.x] = 0.0f;
}

// ---------------------------------------------------------------------------
// Kernel 1: offsets[i] = lower_bound(segment_ids, i); offsets[n] = total
// segment_ids is sorted, so each triple's char range is [offs[i], offs[i+1]).
// 6.4MB array is L2-resident -> ~21 cheap probes per triple.
// ---------------------------------------------------------------------------
__global__ void jl_build_offsets(const int* __restrict__ seg, int total, int n,
                                 int* __restrict__ offs) {
    int i = blockIdx.x * blockDim.x + threadIdx.x;
    if (i > n) return;
    if (i == n) { offs[n] = total; return; }
    int lo = 0, hi = total;
    while (lo < hi) {
        int mid = (lo + hi) >> 1;
        if (seg[mid] < i) lo = mid + 1; else hi = mid;
    }
    offs[i] = lo;
}

// ---------------------------------------------------------------------------
// Kernel 2: fused gather + segment-sum + norms + WMMA row-dots.
// One wave32 per 16-triple tile. T/H tiles staged in LDS; per-triple dots
// computed as diag( T(16x128) * H^T(128x16) ) via 32 chained
// v_wmma_f32_16x16x4_f32 ops. Three scalars merged with f32 global atomics.
// ---------------------------------------------------------------------------
__launch_bounds__(THREADS)
__global__ void jl_fused_main(const float* __restrict__ ce,      // [10000,128] char embeddings
                              const float* __restrict__ ee,      // [N_ENT,128] entity embeddings
                              const int*   __restrict__ head_ids,
                              const int*   __restrict__ char_ids,
                              const int*   __restrict__ offs,    // [n+1]
                              int n,
                              float* __restrict__ scal) {        // [0]=sum(h.t) [1]=sum(h^2) [2]=sum(t^2)
    __shared__ float sT[WAVES_PER_BLOCK][16][ROWP];
    __shared__ float sH[WAVES_PER_BLOCK][16][ROWP];

    const int lane   = threadIdx.x & 31;
    const int w      = threadIdx.x >> 5;
    const int waveId = blockIdx.x * WAVES_PER_BLOCK + w;
    const int nWaves = gridDim.x * WAVES_PER_BLOCK;
    const int numTiles = (n + 15) >> 4;
    const int col = lane * 4;              // each lane owns 4 consecutive dims (float4)

    float sshp = 0.0f, sstp = 0.0f, shtp = 0.0f;

    for (int tile = waveId; tile < numTiles; tile += nWaves) {
        const int base = tile << 4;

        // ---- gather phase: build T and H tiles (uniform control flow per wave) ----
        for (int m = 0; m < 16; ++m) {
            const int tri = base + m;
            float4 acc = make_float4(0.f, 0.f, 0.f, 0.f);
            float4 hv  = make_float4(0.f, 0.f, 0.f, 0.f);
            if (tri < n) {
                const int beg = offs[tri];
                const int end = offs[tri + 1];
                for (int c = beg; c < end; ++c) {
                    const int cid = char_ids[c];   // uniform -> scalarized load
                    const float4 v = *(const float4*)(ce + (size_t)cid * DIM + col);
                    acc.x += v.x; acc.y += v.y; acc.z += v.z; acc.w += v.w;
                }
                const int hid = head_ids[tri];     // uniform
                hv = *(const float4*)(ee + (size_t)hid * DIM + col);
            }
            *(float4*)&sT[w][m][col] = acc;
            *(float4*)&sH[w][m][col] = hv;
            sstp += acc.x * acc.x + acc.y * acc.y + acc.z * acc.z + acc.w * acc.w;
            sshp += hv.x  * hv.x  + hv.y  * hv.y  + hv.z  * hv.z  + hv.w  * hv.w;
        }

        // LDS is wave-private here; DS ops from one wave complete in order.
        // Keep the compiler from reordering reads before the writes above.
        asm volatile("" ::: "memory");
        __builtin_amdgcn_wave_barrier();

        // ---- WMMA phase: D = sum_k T(16x4) x H^T(4x16), accumulate over K=128 ----
        // A (16x4 f32): lanes 0-15 -> M=lane, K=0,1 ; lanes 16-31 -> M=lane-16, K=2,3
        // B (4x16 f32): mirrored with N in place of M.
        v8f cacc = {};
        const int row  = lane & 15;
        const int koff = (lane >> 4) << 1;
#pragma unroll
        for (int kk = 0; kk < 32; ++kk) {
            const int k = (kk << 2) + koff;
            v2f a = *(const v2f*)&sT[w][row][k];
            v2f b = *(const v2f*)&sH[w][row][k];
            cacc = __builtin_amdgcn_wmma_f32_16x16x4_f32(
                /*neg_a=*/false, a, /*neg_b=*/false, b,
                /*c_mod=*/(short)0, cacc, /*reuse_a=*/false, /*reuse_b=*/false);
        }

        // diag(D): C/D layout -> lanes 0-7 hold D[j][j] in VGPR j;
        //                         lanes 24-31 hold D[8+j][8+j] in VGPR j (j = lane-24).
        float d = 0.0f;
#pragma unroll
        for (int j = 0; j < 8; ++j) {
            d += (lane == j)      ? cacc[j] : 0.0f;
            d += (lane == 24 + j) ? cacc[j] : 0.0f;
        }
        shtp += d;

        asm volatile("" ::: "memory");
        __builtin_amdgcn_wave_barrier();   // before next iteration overwrites LDS
    }

    // ---- wave32 reduction of the three partials ----
#pragma unroll
    for (int off = 16; off > 0; off >>= 1) {
        shtp += __shfl_xor(shtp, off, 32);
        sshp += __shfl_xor(sshp, off, 32);
        sstp += __shfl_xor(sstp, off, 32);
    }
    if (lane == 0) {
        atomicAdd(&scal[0], shtp);
        atomicAdd(&scal[1], sshp);
        atomicAdd(&scal[2], sstp);
    }
}

// ---------------------------------------------------------------------------
// Kernel 3: out = N - sum(h.t) / (||H||_F * ||T||_F)
// ---------------------------------------------------------------------------
__global__ void jl_finalize(const float* __restrict__ scal, float* __restrict__ out, int n) {
    if (threadIdx.x == 0 && blockIdx.x == 0) {
        const float denom = sqrtf(scal[1]) * sqrtf(scal[2]);
        out[0] = (float)n - scal[0] / denom;
    }
}

// ---------------------------------------------------------------------------
extern "C" void kernel_launch(void* const* d_in, const int* in_sizes, int n_in,
                              void* d_out, int out_size, void* d_ws, size_t ws_size,
                              hipStream_t stream) {
    const float* ce       = (const float*)d_in[0];  // char_embeddings   [10000*128]
    const float* ee       = (const float*)d_in[1];  // entity_embeddings [100000*128]
    const int*   head_ids = (const int*)  d_in[2];  // [N]
    const int*   char_ids = (const int*)  d_in[3];  // [TOTAL_CHARS]
    const int*   seg_ids  = (const int*)  d_in[4];  // [TOTAL_CHARS], sorted

    const int n     = in_sizes[2];
    const int total = in_sizes[3];

    // workspace: offsets[n+1] ints, then (256B-aligned) 4 scalar floats
    int*   offs = (int*)d_ws;
    size_t offBytes = (((size_t)(n + 1) * sizeof(int)) + 255) & ~(size_t)255;
    float* scal = (float*)((char*)d_ws + offBytes);

    jl_zero_scalars<<<1, 32, 0, stream>>>(scal);

    jl_build_offsets<<<(n + 1 + 255) / 256, 256, 0, stream>>>(seg_ids, total, n, offs);

    const int numTiles = (n + 15) / 16;
    int blocks = (numTiles + WAVES_PER_BLOCK - 1) / WAVES_PER_BLOCK;
    if (blocks > 8192) blocks = 8192;   // grid-stride cap
    jl_fused_main<<<blocks, THREADS, 0, stream>>>(ce, ee, head_ids, char_ids, offs, n, scal);

    jl_finalize<<<1, 1, 0, stream>>>(scal, (float*)d_out, n);
}